// FastAttentionLayer_627065225540
// MI455X (gfx1250) — compile-verified
//
#include <hip/hip_runtime.h>

// ---------------------------------------------------------------------------
// FastAttentionLayer for MI455X (gfx1250, wave32, WMMA + async-LDS pipeline).
// Shapes fixed by reference: B=4, T=12, N=2048, D=256, H=8, hd=32.
// ---------------------------------------------------------------------------

typedef __attribute__((ext_vector_type(2))) float v2f;
typedef __attribute__((ext_vector_type(8))) float v8f;

#define WMMA_F32(a, b, c) \
  __builtin_amdgcn_wmma_f32_16x16x4_f32(false, (a), false, (b), (short)0, (c), false, false)

static __device__ __forceinline__ v8f vzero8() {
  v8f z;
#pragma unroll
  for (int i = 0; i < 8; ++i) z[i] = 0.f;
  return z;
}

// Async global->LDS copy, 16 bytes per lane (GLOBAL_LOAD_ASYNC_TO_LDS_B128,
// tracked by ASYNCcnt).  Generic LDS pointers carry the LDS byte offset in
// their low 32 bits (flat aperture), which is what VDST wants.
static __device__ __forceinline__ void async_copy_b128(const float* g, float* l) {
  const unsigned int ldsoff = (unsigned int)(unsigned long long)l;
  const unsigned long long ga = (unsigned long long)g;
  asm volatile("global_load_async_to_lds_b128 %0, %1, off"
               :: "v"(ldsoff), "v"(ga) : "memory");
}

static __device__ __forceinline__ void wait_async0() {
  asm volatile("s_wait_asynccnt 0x0" ::: "memory");
}

// ---------------------------------------------------------------------------
// C[m,n] = sum_k A[m,k] * W[n,k] (+ bias[n]).  A:[M,K], W:[N,K] row-major.
// Block (8 waves) computes 128x128 of C; K blocked by 64 with double-buffered
// LDS tiles filled by async-to-LDS loads; inner loop is ds_load_b64 -> WMMA.
// Requires M%128==0, N%128==0, K%64==0.
// ---------------------------------------------------------------------------
#define KB 64
#define LDS_STRIDE 68  // KB + 4 pad: conflict-free 2-dword fragment reads

__global__ void gemm_nt_wmma_f32(const float* __restrict__ A,
                                 const float* __restrict__ W,
                                 const float* __restrict__ bias,
                                 float* __restrict__ C,
                                 int M, int N, int K) {
  extern __shared__ float smem[];  // [2][A:128*68 | B:128*68]
  const int tid = threadIdx.x;
  const int lane = tid & 31, wid = tid >> 5;  // 8 waves
  const int ncb = N >> 7;
  const int mblk = blockIdx.x / ncb;
  const int nblk = blockIdx.x - mblk * ncb;
  const int m0 = mblk * 128, n0 = nblk * 128;
  const int row = lane & 15, khalf = lane >> 4;

  // issue one k-block's worth of async tile loads into buffer b
  auto stage = [&](int kb, int b) {
    float* As = smem + b * (2 * 128 * LDS_STRIDE);
    float* Bs = As + 128 * LDS_STRIDE;
#pragma unroll
    for (int t = 0; t < 8; ++t) {
      const int c = tid + t * 256;    // 2048 chunks of 4 floats per tile
      const int r = c >> 4;           // tile row 0..127
      const int kp = (c & 15) * 4;    // 0..60
      async_copy_b128(A + (size_t)(m0 + r) * K + kb + kp, As + r * LDS_STRIDE + kp);
      async_copy_b128(W + (size_t)(n0 + r) * K + kb + kp, Bs + r * LDS_STRIDE + kp);
    }
  };

  v8f acc[8];
#pragma unroll
  for (int t = 0; t < 8; ++t) acc[t] = vzero8();

  stage(0, 0);
  wait_async0();
  __syncthreads();

  const int nkb = K / KB;
  for (int kb = 0; kb < nkb; ++kb) {
    if (kb + 1 < nkb) stage((kb + 1) * KB, (kb + 1) & 1);  // prefetch next
    const float* As = smem + (kb & 1) * (2 * 128 * LDS_STRIDE);
    const float* Bs = As + 128 * LDS_STRIDE;
    const float* arow = As + (wid * 16 + row) * LDS_STRIDE;  // wave's 16 rows
#pragma unroll
    for (int ks = 0; ks < KB / 4; ++ks) {
      const int kk = ks * 4 + 2 * khalf;
      const v2f a = *(const v2f*)(arow + kk);
#pragma unroll
      for (int t = 0; t < 8; ++t) {
        const v2f b = *(const v2f*)(Bs + (t * 16 + row) * LDS_STRIDE + kk);
        acc[t] = WMMA_F32(a, b, acc[t]);
      }
    }
    wait_async0();     // next buffer resident before anyone reads it
    __syncthreads();   // and this buffer fully read before it is overwritten
  }

  const int mw = m0 + wid * 16;
#pragma unroll
  for (int t = 0; t < 8; ++t) {
    const int col = n0 + t * 16 + row;
    const float bv = bias ? bias[col] : 0.f;
#pragma unroll
    for (int r = 0; r < 8; ++r)
      C[(size_t)(mw + r + 8 * khalf) * N + col] = acc[t][r] + bv;  // D: M=r / r+8
  }
}

// ---------------------------------------------------------------------------
// In-place L2 normalize of q and k, per (row, head): 32 contiguous channels.
// ---------------------------------------------------------------------------
__global__ void l2norm_qk_kernel(float* __restrict__ qkv, int Mrows) {
  const int lane = threadIdx.x & 31;
  const int wave = blockIdx.x * (blockDim.x >> 5) + (threadIdx.x >> 5);
  if (wave >= Mrows * 16) return;  // wave-uniform
  const int h    = wave & 7;
  const int qk   = (wave >> 3) & 1;
  const int rowm = wave >> 4;
  const size_t base = (size_t)rowm * 768 + (size_t)qk * 256 + h * 32;
  const float v = qkv[base + lane];
  float s = v * v;
#pragma unroll
  for (int off = 16; off > 0; off >>= 1) s += __shfl_xor(s, off);
  const float scale = 1.f / fmaxf(sqrtf(s), 1e-12f);
  qkv[base + lane] = v * scale;
}

// ---------------------------------------------------------------------------
// Spatial: kvs[bt,h][m][d] = sum_{l<2048} kn[l][m]*v[l][d]; kssum = sum_l kn.
// One block (4 waves) per (bt,h); K split across waves, LDS reduction.
// ---------------------------------------------------------------------------
__global__ void spatial_kvs_kernel(const float* __restrict__ qkv,
                                   float* __restrict__ kvs,
                                   float* __restrict__ kss) {
  __shared__ float lds[4 * 1024 + 4 * 32];
  const int bh = blockIdx.x;
  const int h = bh & 7, bt = bh >> 3;
  const int wid = threadIdx.x >> 5, lane = threadIdx.x & 31;
  const int row = lane & 15, khalf = lane >> 4;
  const size_t base = (size_t)bt * 2048 * 768;
  const float* kp = qkv + base + 256 + h * 32;
  const float* vp = qkv + base + 512 + h * 32;

  v8f acc[2][2];
#pragma unroll
  for (int i = 0; i < 2; ++i)
#pragma unroll
    for (int j = 0; j < 2; ++j) acc[i][j] = vzero8();

  float ks0 = 0.f, ks1 = 0.f;
  const int l0 = wid * 512;
  for (int l = 0; l < 512; l += 4) {
    const int kk = l0 + l + 2 * khalf;
    const float* k0p = kp + (size_t)kk * 768;
    const float* k1p = kp + (size_t)(kk + 1) * 768;
    v2f a0; a0.x = k0p[row];      a0.y = k1p[row];   // A = kn^T
    v2f a1; a1.x = k0p[16 + row]; a1.y = k1p[16 + row];
    ks0 += a0.x + a0.y;
    ks1 += a1.x + a1.y;
    const float* v0p = vp + (size_t)kk * 768;
    const float* v1p = vp + (size_t)(kk + 1) * 768;
    v2f b0; b0.x = v0p[row];      b0.y = v1p[row];   // B = v
    v2f b1; b1.x = v0p[16 + row]; b1.y = v1p[16 + row];
    acc[0][0] = WMMA_F32(a0, b0, acc[0][0]);
    acc[0][1] = WMMA_F32(a0, b1, acc[0][1]);
    acc[1][0] = WMMA_F32(a1, b0, acc[1][0]);
    acc[1][1] = WMMA_F32(a1, b1, acc[1][1]);
  }

  float* slab = lds + wid * 1024;
#pragma unroll
  for (int mt = 0; mt < 2; ++mt)
#pragma unroll
    for (int dt = 0; dt < 2; ++dt)
#pragma unroll
      for (int r = 0; r < 8; ++r)
        slab[(mt * 16 + r + 8 * khalf) * 32 + dt * 16 + row] = acc[mt][dt][r];

  ks0 += __shfl_xor(ks0, 16);
  ks1 += __shfl_xor(ks1, 16);
  if (lane < 16) {
    lds[4096 + wid * 32 + row]      = ks0;
    lds[4096 + wid * 32 + 16 + row] = ks1;
  }
  __syncthreads();

  for (int idx = threadIdx.x; idx < 1024; idx += blockDim.x)
    kvs[(size_t)bh * 1024 + idx] =
        lds[idx] + lds[1024 + idx] + lds[2048 + idx] + lds[3072 + idx];
  if (threadIdx.x < 32)
    kss[(size_t)bh * 32 + threadIdx.x] =
        lds[4096 + threadIdx.x] + lds[4096 + 32 + threadIdx.x] +
        lds[4096 + 64 + threadIdx.x] + lds[4096 + 96 + threadIdx.x];
}

// ---------------------------------------------------------------------------
// Spatial output: out_s = (qn @ kvs + L*v) / (qn . kssum + L), L = 2048.
// ---------------------------------------------------------------------------
__global__ void spatial_out_kernel(const float* __restrict__ qkv,
                                   const float* __restrict__ kvs,
                                   const float* __restrict__ kss,
                                   float* __restrict__ cat) {
  const int lane = threadIdx.x & 31;
  const int wave = blockIdx.x * (blockDim.x >> 5) + (threadIdx.x >> 5);
  const int ltile = wave & 127;
  const int h     = (wave >> 7) & 7;
  const int bt    = wave >> 10;
  const int row = lane & 15, khalf = lane >> 4;
  const int l0 = ltile * 16;
  const size_t base = (size_t)bt * 2048 * 768;
  const float* qp   = qkv + base + h * 32;
  const float* vp   = qkv + base + 512 + h * 32;
  const float* kvsp = kvs + (size_t)(bt * 8 + h) * 1024;
  const float* kssp = kss + (size_t)(bt * 8 + h) * 32;

  v8f acc[2]; acc[0] = vzero8(); acc[1] = vzero8();
  const float* qrow = qp + (size_t)(l0 + row) * 768;
  for (int k = 0; k < 32; k += 4) {
    const int kk = k + 2 * khalf;
    v2f a; a.x = qrow[kk]; a.y = qrow[kk + 1];
    v2f b0; b0.x = kvsp[kk * 32 + row];      b0.y = kvsp[(kk + 1) * 32 + row];
    v2f b1; b1.x = kvsp[kk * 32 + 16 + row]; b1.y = kvsp[(kk + 1) * 32 + 16 + row];
    acc[0] = WMMA_F32(a, b0, acc[0]);
    acc[1] = WMMA_F32(a, b1, acc[1]);
  }

  float pd = 0.f;
#pragma unroll
  for (int m = 0; m < 16; ++m) pd += qrow[khalf * 16 + m] * kssp[khalf * 16 + m];
  pd += __shfl_xor(pd, 16);  // lane holds denom-dot for row (lane&15)

  const float Lf = 2048.f;
#pragma unroll
  for (int t = 0; t < 2; ++t) {
    const int d = t * 16 + row;
#pragma unroll
    for (int r = 0; r < 8; ++r) {
      const int l = l0 + r + 8 * khalf;
      const float den = __shfl(pd, r + 8 * khalf) + Lf;
      const float vv = vp[(size_t)l * 768 + d];
      cat[(size_t)(bt * 2048 + l) * 512 + h * 32 + d] = (acc[t][r] + Lf * vv) / den;
    }
  }
}

// ---------------------------------------------------------------------------
// Temporal attention fused per (b,n): L = T = 12 (VALU+LDS is right at this
// size).  One block per (b,n), one wave per head.
// ---------------------------------------------------------------------------
__global__ void temporal_kernel(const float* __restrict__ qkv,
                                float* __restrict__ cat) {
  __shared__ float q_s[8][12][32];
  __shared__ float k_s[8][12][32];
  __shared__ float v_s[8][12][32];
  __shared__ float ks_s[8][32];
  const int h = threadIdx.x >> 5;
  const int lane = threadIdx.x & 31;
  const int b = blockIdx.x >> 11;
  const int n = blockIdx.x & 2047;

#pragma unroll
  for (int t = 0; t < 12; ++t) {
    const size_t rb = ((size_t)(b * 12 + t) * 2048 + n) * 768 + h * 32 + lane;
    q_s[h][t][lane] = qkv[rb];
    k_s[h][t][lane] = qkv[rb + 256];
    v_s[h][t][lane] = qkv[rb + 512];
  }
  __syncthreads();

  float kv[32];
#pragma unroll
  for (int m = 0; m < 32; ++m) kv[m] = 0.f;
#pragma unroll
  for (int t = 0; t < 12; ++t) {
    const float vv = v_s[h][t][lane];
#pragma unroll
    for (int m = 0; m < 32; ++m) kv[m] += k_s[h][t][m] * vv;
  }
  float ks = 0.f;
#pragma unroll
  for (int t = 0; t < 12; ++t) ks += k_s[h][t][lane];
  ks_s[h][lane] = ks;
  __syncthreads();

  const float Lf = 12.f;
#pragma unroll
  for (int t = 0; t < 12; ++t) {
    float num = 0.f, den = 0.f;
#pragma unroll
    for (int m = 0; m < 32; ++m) {
      const float qm = q_s[h][t][m];
      num += qm * kv[m];
      den += qm * ks_s[h][m];
    }
    const float o = (num + Lf * v_s[h][t][lane]) / (den + Lf);
    cat[((size_t)(b * 12 + t) * 2048 + n) * 512 + 256 + h * 32 + lane] = o;
  }
}

// ---------------------------------------------------------------------------
// Launch.
// ---------------------------------------------------------------------------
extern "C" void kernel_launch(void* const* d_in, const int* in_sizes, int n_in,
                              void* d_out, int out_size, void* d_ws, size_t ws_size,
                              hipStream_t stream) {
  (void)in_sizes; (void)n_in; (void)out_size; (void)ws_size;
  const float* x    = (const float*)d_in[0];
  const float* Wqkv = (const float*)d_in[1];
  const float* Wout = (const float*)d_in[2];
  const float* bout = (const float*)d_in[3];
  float* out = (float*)d_out;

  const int M = 4 * 12 * 2048;  // 98304 rows
  float* ws   = (float*)d_ws;
  float* qkv  = ws;                           // [M, 768]
  float* cat  = qkv + (size_t)M * 768;        // [M, 512]
  float* kvs  = cat + (size_t)M * 512;        // [48*8, 32, 32]
  float* kss  = kvs + (size_t)48 * 8 * 1024;  // [48*8, 32]

  const size_t gemm_lds = (size_t)2 * 2 * 128 * LDS_STRIDE * sizeof(float);  // 136 KB

  // 1) qkv = x @ Wqkv^T
  gemm_nt_wmma_f32<<<(M / 128) * (768 / 128), 256, gemm_lds, stream>>>(
      x, Wqkv, nullptr, qkv, M, 768, 256);
  // 2) L2-normalize q, k per head (in place)
  l2norm_qk_kernel<<<(M * 16) / 8, 256, 0, stream>>>(qkv, M);
  // 3) spatial kvs / ks_sum
  spatial_kvs_kernel<<<48 * 8, 128, 0, stream>>>(qkv, kvs, kss);
  // 4) spatial out -> cat[:, :256]
  spatial_out_kernel<<<(48 * 8 * 128) / 4, 128, 0, stream>>>(qkv, kvs, kss, cat);
  // 5) temporal attention -> cat[:, 256:]
  temporal_kernel<<<4 * 2048, 256, 0, stream>>>(qkv, cat);
  // 6) out = cat @ Wout^T + bout
  gemm_nt_wmma_f32<<<(M / 128) * (256 / 128), 256, gemm_lds, stream>>>(
      cat, Wout, bout, out, M, 256, 512);
}